// AttentionDecoderRNN_35304631173858
// MI455X (gfx1250) — compile-verified
//
#include <hip/hip_runtime.h>
#include <hip/hip_bf16.h>
#include <math.h>

// Problem constants (S, B, H) = (512, 128, 1024)
#define SDIM 512
#define BDIM 128
#define HDIM 1024
#define N3H  3072
#define NPART 8           // attention partial blocks per batch row
#define LDA  516          // padded LDS row stride (floats); 516 % 64 == 4 -> conflict-free WMMA reads
#define L2KEEP 320        // s < L2KEEP kept temporal in L2 (~160MB resident); rest streamed NT

typedef float v2f  __attribute__((ext_vector_type(2)));
typedef float v8f  __attribute__((ext_vector_type(8)));
typedef float f4v  __attribute__((ext_vector_type(4)));

// ---------------------------------------------------------------------------
// GRU gate GEMM:  out[z] = A[z] @ W[z]^T + bias[z]
//   z==0: gx = x @ W_ih^T + b_ih    (x is all-zero at t==0 -> bias only)
//   z==1: gh = h @ W_hh^T + b_hh
// C = [128, 3072], K = 1024. One 16x16 D-tile per wave, fp32 WMMA 16x16x4
// (fp32 matrix pipe keeps the 512-step recurrence numerically exact).
// ---------------------------------------------------------------------------
__global__ __launch_bounds__(256) void gru_gates_gemm(
    const float* __restrict__ Ax, const float* __restrict__ Ah,
    const float* __restrict__ Wih, const float* __restrict__ Whh,
    const float* __restrict__ bih, const float* __restrict__ bhh,
    float* __restrict__ gx, float* __restrict__ gh, int x_is_zero)
{
    __shared__ float As[16 * LDA];                 // 16 x 512 staged K-slab, padded
    const int z = blockIdx.z;
    const float* A    = z ? Ah  : Ax;
    const float* W    = z ? Whh : Wih;
    const float* bias = z ? bhh : bih;
    float*       out  = z ? gh  : gx;

    const int mbase = blockIdx.x * 16;             // 8 m-tiles
    const int tid   = threadIdx.x;
    const int wave  = tid >> 5;
    const int lane  = tid & 31;
    const int hi    = lane >> 4;                   // 0: lanes 0-15, 1: lanes 16-31
    const int lo    = lane & 15;
    const int nbase = (blockIdx.y * 8 + wave) * 16;  // 24*8 = 192 n-tiles

    v8f c = {};
    const bool skip = (z == 0) && (x_is_zero != 0);
    if (!skip) {
        const float* wrow = W + (size_t)(nbase + lo) * HDIM;   // L2-resident weight row
        for (int kk = 0; kk < HDIM; kk += 512) {
            // cooperative stage of A[16][512] into LDS (float4, coalesced)
            #pragma unroll
            for (int i = 0; i < 8; ++i) {
                int fi  = tid + 256 * i;          // float4 index 0..2047
                int row = fi >> 7;                // 128 float4 per row
                int c4  = fi & 127;
                const f4v v = *(const f4v*)(A + (size_t)(mbase + row) * HDIM + kk + c4 * 4);
                *(f4v*)(&As[row * LDA + c4 * 4]) = v;
            }
            __syncthreads();
            #pragma unroll 8
            for (int k0 = 0; k0 < 512; k0 += 4) {
                const int k = k0 + 2 * hi;
                v2f a, b;
                a.x = As[lo * LDA + k];
                a.y = As[lo * LDA + k + 1];
                const float2 bw = *(const float2*)(wrow + kk + k);
                b.x = bw.x; b.y = bw.y;
                c = __builtin_amdgcn_wmma_f32_16x16x4_f32(
                        false, a, false, b, (short)0, c, false, false);
            }
            __syncthreads();
        }
    }
    const int col = nbase + lo;
    const float bv = bias[col];
    #pragma unroll
    for (int i = 0; i < 8; ++i) {
        const int row = mbase + i + 8 * hi;       // C layout: VGPR i -> M = i + 8*hi
        out[(size_t)row * N3H + col] = c[i] + bv;
    }
}

// ---------------------------------------------------------------------------
// GRU nonlinearity: h_new = (1-z)*tanh(xn + r*hn) + z*h,  r,z = sigmoid sums
// ---------------------------------------------------------------------------
__global__ __launch_bounds__(256) void gru_gate_act(
    const float* __restrict__ gx, const float* __restrict__ gh,
    const float* __restrict__ hprev, float* __restrict__ hnew)
{
    const int idx = blockIdx.x * 256 + threadIdx.x;   // 0 .. B*H-1
    const int b = idx >> 10;
    const int j = idx & 1023;
    const float* gxb = gx + (size_t)b * N3H;
    const float* ghb = gh + (size_t)b * N3H;
    const float r  = 1.f / (1.f + __expf(-(gxb[j] + ghb[j])));
    const float zz = 1.f / (1.f + __expf(-(gxb[HDIM + j] + ghb[HDIM + j])));
    const float n  = tanhf(gxb[2 * HDIM + j] + r * ghb[2 * HDIM + j]);
    const float hp = hprev[idx];
    hnew[idx] = (1.f - zz) * n + zz * hp;
}

// ---------------------------------------------------------------------------
// Attention pass 1: flash-style single pass over enc (the HBM-bound part).
// Block (b, part): 8 waves; wave handles s = part*8+wave + 64*i, s < len[b].
// Cache policy: s < L2KEEP rows load temporal (stay L2-resident across all
// 512 steps, ~160MB); s >= L2KEEP rows load non-temporal (stream from HBM
// without evicting the resident slice). This breaks the cyclic-LRU thrash of
// re-reading a 256MB tensor through a 192MB L2 every step.
// ---------------------------------------------------------------------------
__global__ __launch_bounds__(256) void attn_partial(
    const float* __restrict__ enc, const float* __restrict__ hnew,
    const int* __restrict__ lengths,
    float* __restrict__ pm, float* __restrict__ pl, float* __restrict__ pctx)
{
    const int b    = blockIdx.x;
    const int part = blockIdx.y;
    const int tid  = threadIdx.x;
    const int wave = tid >> 5;
    const int lane = tid & 31;
    const int len  = lengths[b];

    // cache h_new[b,:] in registers: lane covers h = j*128 + lane*4 .. +3
    f4v hreg[8];
    const f4v* h4 = (const f4v*)(hnew + (size_t)b * HDIM);
    #pragma unroll
    for (int j = 0; j < 8; ++j) hreg[j] = h4[j * 32 + lane];

    float m = -INFINITY, l = 0.f;
    f4v acc[8];
    #pragma unroll
    for (int j = 0; j < 8; ++j) acc[j] = (f4v){0.f, 0.f, 0.f, 0.f};

    const int slot = part * 8 + wave;             // 0..63
    for (int s = slot; s < len; s += 64) {
        const f4v* e4 = (const f4v*)(enc + ((size_t)s * BDIM + b) * HDIM);
        // cover latency of this wave's next s-row (global_prefetch_b8)
        if (s + 64 < len)
            __builtin_prefetch(enc + ((size_t)(s + 64) * BDIM + b) * HDIM + lane * 16, 0, 0);
        f4v ev[8];
        if (s < L2KEEP) {
            #pragma unroll
            for (int j = 0; j < 8; ++j) ev[j] = e4[j * 32 + lane];
        } else {
            #pragma unroll
            for (int j = 0; j < 8; ++j) ev[j] = __builtin_nontemporal_load(e4 + j * 32 + lane);
        }
        float p = 0.f;
        #pragma unroll
        for (int j = 0; j < 8; ++j) {
            const f4v tprod = ev[j] * hreg[j];
            p += tprod.x + tprod.y + tprod.z + tprod.w;
        }
        #pragma unroll
        for (int off = 16; off > 0; off >>= 1) p += __shfl_xor(p, off, 32);
        const float mn   = fmaxf(m, p);
        const float cold = __expf(m - mn);
        const float w    = __expf(p - mn);
        l = l * cold + w;
        #pragma unroll
        for (int j = 0; j < 8; ++j) acc[j] = acc[j] * cold + w * ev[j];
        m = mn;
    }

    // combine the 8 waves of this block in LDS
    __shared__ float sm[8], sl[8];
    __shared__ float sctx[8][HDIM];               // 32 KB
    #pragma unroll
    for (int j = 0; j < 8; ++j) ((f4v*)sctx[wave])[j * 32 + lane] = acc[j];
    if (lane == 0) { sm[wave] = m; sl[wave] = l; }
    __syncthreads();

    float M = sm[0];
    #pragma unroll
    for (int w2 = 1; w2 < 8; ++w2) M = fmaxf(M, sm[w2]);
    float coef[8]; float L = 0.f;
    #pragma unroll
    for (int w2 = 0; w2 < 8; ++w2) {
        const float cc = (sl[w2] > 0.f) ? __expf(sm[w2] - M) : 0.f;  // guard -inf - -inf
        coef[w2] = cc; L += cc * sl[w2];
    }
    // each thread combines 4 h positions
    f4v r = (f4v){0.f, 0.f, 0.f, 0.f};
    #pragma unroll
    for (int w2 = 0; w2 < 8; ++w2) r += coef[w2] * ((const f4v*)sctx[w2])[tid];
    *(f4v*)(pctx + ((size_t)b * NPART + part) * HDIM + tid * 4) = r;
    if (tid == 0) { pm[b * NPART + part] = M; pl[b * NPART + part] = L; }
}

// ---------------------------------------------------------------------------
// Attention finalize + projection: combine NPART partials -> context, then
// p = W_p @ [context; h_new] + b_p, written to out[b, t, 0:2].
// ---------------------------------------------------------------------------
__global__ __launch_bounds__(256) void attn_finalize(
    const float* __restrict__ pm, const float* __restrict__ pl,
    const float* __restrict__ pctx, const float* __restrict__ hnew,
    const float* __restrict__ Wp, const float* __restrict__ bp,
    float* __restrict__ out, int t)
{
    const int b = blockIdx.x, tid = threadIdx.x;
    float M = -INFINITY;
    #pragma unroll
    for (int p = 0; p < NPART; ++p) M = fmaxf(M, pm[b * NPART + p]);
    float coef[NPART]; float L = 0.f;
    #pragma unroll
    for (int p = 0; p < NPART; ++p) {
        const float lp = pl[b * NPART + p];
        const float cc = (lp > 0.f) ? __expf(pm[b * NPART + p] - M) : 0.f;
        coef[p] = cc; L += cc * lp;
    }
    const float invL = 1.f / L;                   // len[b] >= 1 -> L > 0

    __shared__ float ctx[HDIM];
    f4v r = (f4v){0.f, 0.f, 0.f, 0.f};
    #pragma unroll
    for (int p = 0; p < NPART; ++p)
        r += coef[p] * (*(const f4v*)(pctx + ((size_t)b * NPART + p) * HDIM + tid * 4));
    r *= invL;
    *(f4v*)(ctx + tid * 4) = r;
    __syncthreads();

    __shared__ float red[8];
    const float* hb = hnew + (size_t)b * HDIM;
    for (int i = 0; i < 2; ++i) {
        float partl = 0.f;
        for (int h = tid; h < HDIM; h += 256)
            partl += Wp[i * 2048 + h] * ctx[h] + Wp[i * 2048 + HDIM + h] * hb[h];
        #pragma unroll
        for (int off = 16; off > 0; off >>= 1) partl += __shfl_xor(partl, off, 32);
        if ((tid & 31) == 0) red[tid >> 5] = partl;
        __syncthreads();
        if (tid == 0) {
            float s = 0.f;
            #pragma unroll
            for (int w = 0; w < 8; ++w) s += red[w];
            out[((size_t)b * SDIM + t) * 2 + i] = s + bp[i];
        }
        __syncthreads();
    }
}

// ---------------------------------------------------------------------------
extern "C" void kernel_launch(void* const* d_in, const int* in_sizes, int n_in,
                              void* d_out, int out_size, void* d_ws, size_t ws_size,
                              hipStream_t stream)
{
    const float* enc   = (const float*)d_in[0];   // [S,B,H]
    const float* h0in  = (const float*)d_in[1];   // [1,B,H]
    const float* Wih   = (const float*)d_in[2];   // [3H,H]
    const float* Whh   = (const float*)d_in[3];   // [3H,H]
    const float* bih   = (const float*)d_in[4];   // [3H]
    const float* bhh   = (const float*)d_in[5];   // [3H]
    const float* Wp    = (const float*)d_in[6];   // [2,2H]
    const float* bp    = (const float*)d_in[7];   // [2]
    const int*   lens  = (const int*)d_in[8];     // [B]
    float*       out   = (float*)d_out;           // [B,S,2]

    // workspace carve-up (floats)
    float* w    = (float*)d_ws;
    float* gx   = w;                      w += (size_t)BDIM * N3H;       // 393216
    float* gh   = w;                      w += (size_t)BDIM * N3H;
    float* hb0  = w;                      w += (size_t)BDIM * HDIM;
    float* hb1  = w;                      w += (size_t)BDIM * HDIM;
    float* pm   = w;                      w += (size_t)BDIM * NPART;
    float* pl   = w;                      w += (size_t)BDIM * NPART;
    float* pctx = w;                      w += (size_t)BDIM * NPART * HDIM;
    float* hbuf[2] = { hb0, hb1 };

    for (int t = 0; t < SDIM; ++t) {
        // reference feeds h_new back as next x, so x == h for t >= 1
        const float* hin = (t == 0) ? h0in : hbuf[(t - 1) & 1];
        float* hout = hbuf[t & 1];
        gru_gates_gemm<<<dim3(8, 24, 2), 256, 0, stream>>>(
            hin, hin, Wih, Whh, bih, bhh, gx, gh, (t == 0) ? 1 : 0);
        gru_gate_act<<<dim3((BDIM * HDIM) / 256), 256, 0, stream>>>(gx, gh, hin, hout);
        attn_partial<<<dim3(BDIM, NPART), 256, 0, stream>>>(enc, hout, lens, pm, pl, pctx);
        attn_finalize<<<dim3(BDIM), 256, 0, stream>>>(pm, pl, pctx, hout, Wp, bp, out, t);
    }
}